// diff_rc2lpc_79336635892365
// MI455X (gfx1250) — compile-verified
//
#include <hip/hip_runtime.h>
#include <stdint.h>

// rc2lpc: Levinson recursion, 2,097,152 independent rows of 16 f32.
// Bandwidth-bound: 268 MB total traffic @ 23.3 TB/s ~= 11.5 us floor; the
// 0.5 GFLOP f32 recursion is noise. Kernel shape = pure streaming:
//   in : global -> LDS via CDNA5 global_load_async_to_lds_b128 (NT hint),
//        lane-linear 512 B/wave/issue, padded LDS tile (80 B row stride)
//   mid: fully unrolled 120-FMA register recursion per row
//   out: results -> LDS -> global via global_store_async_from_lds_b128,
//        lane-linear, guaranteed full 128 B line writes (no RMW at HBM)

#define LPC_N 16
#define ROWS_PER_BLOCK 256
#define ROW_PAD_DW 20          // 80 B row stride: 16B-aligned, bank-spread

typedef float f32x4 __attribute__((ext_vector_type(4)));

__global__ __launch_bounds__(ROWS_PER_BLOCK)
void rc2lpc_async_kernel(const float* __restrict__ rc,
                         float* __restrict__ out,
                         int nrows) {
    __shared__ float smem[ROWS_PER_BLOCK * ROW_PAD_DW];   // 20,480 B

    const int tid     = threadIdx.x;
    const int rowBase = blockIdx.x * ROWS_PER_BLOCK;
    const bool full   = (rowBase + ROWS_PER_BLOCK) <= nrows;

    // Low 32 bits of a flat shared pointer == wave-relative LDS byte address.
    const uint32_t lds_base = (uint32_t)(uintptr_t)(&smem[0]);

    // ---- Stage 1: async copy tile (256 rows x 64 B) global -> LDS ----
    const uint64_t gin =
        (uint64_t)(uintptr_t)rc + (uint64_t)rowBase * (LPC_N * 4);
    if (full) {
#pragma unroll
        for (int j = 0; j < 4; ++j) {
            const int c = j * ROWS_PER_BLOCK + tid;   // 16B chunk 0..1023
            const int r = c >> 2;
            const int p = c & 3;
            const uint64_t gaddr = gin + (uint64_t)c * 16u;
            const uint32_t laddr =
                lds_base + (uint32_t)(r * (ROW_PAD_DW * 4) + p * 16);
            asm volatile(
                "global_load_async_to_lds_b128 %0, %1, off th:TH_LOAD_NT"
                :: "v"(laddr), "v"(gaddr) : "memory");
        }
    } else {
#pragma unroll
        for (int j = 0; j < 4; ++j) {
            const int c = j * ROWS_PER_BLOCK + tid;
            const int r = c >> 2;
            const int p = c & 3;
            if (rowBase + r < nrows) {
                const uint64_t gaddr = gin + (uint64_t)c * 16u;
                const uint32_t laddr =
                    lds_base + (uint32_t)(r * (ROW_PAD_DW * 4) + p * 16);
                asm volatile(
                    "global_load_async_to_lds_b128 %0, %1, off th:TH_LOAD_NT"
                    :: "v"(laddr), "v"(gaddr) : "memory");
            }
        }
    }
    asm volatile("s_wait_asynccnt 0" ::: "memory");
    __syncthreads();

    // ---- Stage 2: per-row Levinson recursion in registers ----
    // Each thread reads and later rewrites ONLY its own LDS row slot, so no
    // extra barrier is needed for the in-place tile reuse.
    if (full || (rowBase + tid) < nrows) {
        f32x4* slot = (f32x4*)(&smem[tid * ROW_PAD_DW]);
        const f32x4 v0 = slot[0], v1 = slot[1], v2 = slot[2], v3 = slot[3];
        const float r0[LPC_N] = { v0.x, v0.y, v0.z, v0.w,
                                  v1.x, v1.y, v1.z, v1.w,
                                  v2.x, v2.y, v2.z, v2.w,
                                  v3.x, v3.y, v3.z, v3.w };
        float l[LPC_N];
        l[0] = r0[0];
#pragma unroll
        for (int i = 1; i < LPC_N; ++i) {
            const float k = r0[i];
            float nl[LPC_N];
#pragma unroll
            for (int j = 0; j < LPC_N; ++j)
                if (j < i) nl[j] = __builtin_fmaf(k, l[i - 1 - j], l[j]);
#pragma unroll
            for (int j = 0; j < LPC_N; ++j)
                if (j < i) l[j] = nl[j];
            l[i] = k;
        }
        slot[0] = (f32x4){ l[0],  l[1],  l[2],  l[3]  };
        slot[1] = (f32x4){ l[4],  l[5],  l[6],  l[7]  };
        slot[2] = (f32x4){ l[8],  l[9],  l[10], l[11] };
        slot[3] = (f32x4){ l[12], l[13], l[14], l[15] };
    }
    __syncthreads();   // all rows' results resident in LDS

    // ---- Stage 3: async store LDS -> global, lane-linear (full lines) ----
    const uint64_t gout =
        (uint64_t)(uintptr_t)out + (uint64_t)rowBase * (LPC_N * 4);
    if (full) {
#pragma unroll
        for (int j = 0; j < 4; ++j) {
            const int c = j * ROWS_PER_BLOCK + tid;
            const int r = c >> 2;
            const int p = c & 3;
            const uint64_t gaddr = gout + (uint64_t)c * 16u;
            const uint32_t laddr =
                lds_base + (uint32_t)(r * (ROW_PAD_DW * 4) + p * 16);
            asm volatile(
                "global_store_async_from_lds_b128 %0, %1, off"
                :: "v"(gaddr), "v"(laddr) : "memory");
        }
    } else {
#pragma unroll
        for (int j = 0; j < 4; ++j) {
            const int c = j * ROWS_PER_BLOCK + tid;
            const int r = c >> 2;
            const int p = c & 3;
            if (rowBase + r < nrows) {
                const uint64_t gaddr = gout + (uint64_t)c * 16u;
                const uint32_t laddr =
                    lds_base + (uint32_t)(r * (ROW_PAD_DW * 4) + p * 16);
                asm volatile(
                    "global_store_async_from_lds_b128 %0, %1, off"
                    :: "v"(gaddr), "v"(laddr) : "memory");
            }
        }
    }
    // S_ENDPGM performs an implicit wait-idle, but make completion explicit:
    asm volatile("s_wait_asynccnt 0" ::: "memory");
}

extern "C" void kernel_launch(void* const* d_in, const int* in_sizes, int n_in,
                              void* d_out, int out_size, void* d_ws, size_t ws_size,
                              hipStream_t stream) {
    const float* rc  = (const float*)d_in[0];
    float*       out = (float*)d_out;

    const int nrows  = in_sizes[0] / LPC_N;                            // 2,097,152
    const int blocks = (nrows + ROWS_PER_BLOCK - 1) / ROWS_PER_BLOCK;  // 8192

    rc2lpc_async_kernel<<<blocks, ROWS_PER_BLOCK, 0, stream>>>(rc, out, nrows);
}